// Graph_to_Featuremaps_6880537608449
// MI455X (gfx1250) — compile-verified
//
#include <hip/hip_runtime.h>
#include <math.h>

// ---------------------------------------------------------------------------
// Graph_to_Featuremaps, MI455X (gfx1250).
//
// Key identity: softmax over the node axis of (score_pix[b,p] + score_node[b,n])
// is invariant to score_pix (constant along the softmax axis), so the attention
// and hence the output are independent of the pixel index p. res_feature is
// mathematically dead. The problem reduces to:
//   a[b,:]    = softmax(x[b] @ nf_node)              (16x7)
//   A'[b,h]   = sum_n a[b,n] * x[b,n,h]              (16x256)
//   mixed     = relu(A' @ weight)                    (16x512, WMMA f32 16x16x4)
//   out[b,o,:] = broadcast mixed[b,o] over 96*96     (302 MB of NT b128 stores)
// Store bandwidth (302 MB @ 23.3 TB/s ~ 13 us) is the only meaningful cost.
// ---------------------------------------------------------------------------

typedef __attribute__((ext_vector_type(2))) float v2f;
typedef __attribute__((ext_vector_type(4))) float f32x4;
typedef __attribute__((ext_vector_type(8))) float v8f;

#define BATCH 16
#define NODES 7
#define HID   256
#define OUTC  512
#define HW    (96 * 96)

// ---- Stage 1: node scores -> softmax -> attention-folded A' (16 x 256) ----
__global__ void prep_kernel(const float* __restrict__ x,        // (16,7,256)
                            const float* __restrict__ node_fea, // (768,1)
                            float* __restrict__ Aprime)         // ws: 16*256
{
    __shared__ float s_attn[BATCH * NODES];
    __shared__ float s_score[BATCH * NODES];
    const int t = threadIdx.x;

    // 112 dot products of length 256 (trivial work).
    if (t < BATCH * NODES) {
        const float* nf = node_fea + OUTC;   // nf_node = node_fea[C:]
        const float* xr = x + t * HID;
        float acc = 0.f;
        for (int h = 0; h < HID; ++h) acc = fmaf(xr[h], nf[h], acc);
        s_score[t] = acc;
    }
    __syncthreads();

    // Per-batch softmax over the 7 nodes.
    if (t < BATCH) {
        float mx = -INFINITY;
        for (int n = 0; n < NODES; ++n) mx = fmaxf(mx, s_score[t * NODES + n]);
        float e[NODES];
        float sum = 0.f;
        for (int n = 0; n < NODES; ++n) {
            e[n] = expf(s_score[t * NODES + n] - mx);
            sum += e[n];
        }
        float inv = 1.f / sum;
        for (int n = 0; n < NODES; ++n) s_attn[t * NODES + n] = e[n] * inv;
    }
    __syncthreads();

    // A'[b,h] = sum_n a[b,n] * x[b,n,h]
    for (int i = t; i < BATCH * HID; i += blockDim.x) {
        const int b = i >> 8;
        const int h = i & (HID - 1);
        float acc = 0.f;
        for (int n = 0; n < NODES; ++n)
            acc = fmaf(s_attn[b * NODES + n], x[(b * NODES + n) * HID + h], acc);
        Aprime[i] = acc;
    }
}

// ---- Stage 2: mixed = relu(A' @ weight) via V_WMMA_F32_16X16X4_F32 ----
// One wave (32 lanes) per 16-wide N tile; M=16 is exactly one tile; K=256 in
// 64 steps of 4. Fragment layouts per ISA 7.12.2 (32-bit 16x4 A / 4x16 B):
//   VGPR0: lanes 0-15 -> K=k0+0, lanes 16-31 -> K=k0+2;  VGPR1: +1.
__global__ void gemm_wmma_kernel(const float* __restrict__ Aprime, // 16x256
                                 const float* __restrict__ W,      // 256x512
                                 float* __restrict__ mixed)        // 16x512
{
    const int lane = threadIdx.x;      // 0..31, wave32
    const int half = lane >> 4;        // 0: lanes 0-15, 1: lanes 16-31
    const int lm   = lane & 15;
    const int ncol = blockIdx.x * 16 + lm;

    v8f c = {};
    for (int k0 = 0; k0 < HID; k0 += 4) {
        const int ka = k0 + 2 * half;
        v2f a, b;
        a.x = Aprime[lm * HID + ka];
        a.y = Aprime[lm * HID + ka + 1];
        b.x = W[ka * OUTC + ncol];
        b.y = W[(ka + 1) * OUTC + ncol];
        // D = A(16x4,f32) * B(4x16,f32) + C   (exact fp32 path)
        c = __builtin_amdgcn_wmma_f32_16x16x4_f32(
                /*neg_a=*/false, a, /*neg_b=*/false, b,
                /*c_mod=*/(short)0, c, /*reuse_a=*/false, /*reuse_b=*/false);
    }

    // C/D layout: VGPR r -> M = r (lanes 0-15) / r+8 (lanes 16-31). Fuse ReLU.
    for (int r = 0; r < 8; ++r) {
        const int m = r + 8 * half;
        mixed[m * OUTC + ncol] = fmaxf(c[r], 0.f);
    }
}

// ---- Stage 3: broadcast mixed[b,o] over 96*96 pixels (302 MB NT stream) ----
__global__ void bcast_kernel(const float* __restrict__ mixed, // 16*512
                             float* __restrict__ out)         // (16,512,96,96)
{
    const int bc = blockIdx.x;               // b*512 + o, uniform per block
    const float v = mixed[bc];
    f32x4 vv = {v, v, v, v};
    f32x4* o = reinterpret_cast<f32x4*>(out) + (size_t)bc * (HW / 4);
    // Output (302 MB) > L2 (192 MB) and never re-read: non-temporal b128 stores.
    for (int i = threadIdx.x; i < HW / 4; i += blockDim.x)
        __builtin_nontemporal_store(vv, o + i);
}

extern "C" void kernel_launch(void* const* d_in, const int* in_sizes, int n_in,
                              void* d_out, int out_size, void* d_ws, size_t ws_size,
                              hipStream_t stream) {
    const float* x        = (const float*)d_in[0];  // input (1,16,7,256)
    // d_in[1] = res_feature: mathematically dead (see header comment), never read.
    const float* node_fea = (const float*)d_in[2];  // (768,1)
    const float* weight   = (const float*)d_in[3];  // (256,512)
    float* out = (float*)d_out;

    float* wsf    = (float*)d_ws;
    float* Aprime = wsf;                 // 16*256 floats
    float* mixed  = wsf + BATCH * HID;   // 16*512 floats

    prep_kernel<<<1, 1024, 0, stream>>>(x, node_fea, Aprime);
    gemm_wmma_kernel<<<OUTC / 16, 32, 0, stream>>>(Aprime, weight, mixed);
    bcast_kernel<<<BATCH * OUTC, 256, 0, stream>>>(mixed, out);
}